// GraphSAGEModel_47571057770997
// MI455X (gfx1250) — compile-verified
//
#include <hip/hip_runtime.h>
#include <stdint.h>

#define N_NODES 100000
#define N_EDGES 1600000
#define D_IN    64
#define D_HID   64
#define D_OUT   32

typedef __attribute__((ext_vector_type(2))) float v2f;
typedef __attribute__((ext_vector_type(8))) float v8f;

// ---------------------------------------------------------------------------
// Utility: zero a float buffer
// ---------------------------------------------------------------------------
__global__ void zero_f32(float* __restrict__ p, int n) {
    int i = blockIdx.x * blockDim.x + threadIdx.x;
    if (i < n) p[i] = 0.0f;
}

// ---------------------------------------------------------------------------
// Degree: deg[dst] += 1 per edge (f32 atomics, resolved in L2)
// ---------------------------------------------------------------------------
__global__ void degree_kernel(const long long* __restrict__ ei,
                              float* __restrict__ deg) {
    int e = blockIdx.x * blockDim.x + threadIdx.x;
    if (e < N_EDGES) {
        int dst = (int)ei[N_EDGES + e];
        atomicAdd(&deg[dst], 1.0f);
    }
}

// rdeg[i] = 1 / max(deg[i], 1)
__global__ void rdeg_kernel(const float* __restrict__ deg,
                            float* __restrict__ rdeg) {
    int i = blockIdx.x * blockDim.x + threadIdx.x;
    if (i < N_NODES) rdeg[i] = 1.0f / fmaxf(deg[i], 1.0f);
}

// ---------------------------------------------------------------------------
// Scatter-add: summed[dst, :] += feat[src, :]   (64 floats per edge)
// One thread handles 4 consecutive floats (float4 gather + 4 f32 atomics).
// Both feat (25.6MB) and summed (25.6MB) fit in the 192MB L2, so the atomic
// traffic resolves in L2, not HBM.
// ---------------------------------------------------------------------------
__global__ void scatter_kernel(const long long* __restrict__ ei,
                               const float* __restrict__ feat,
                               float* __restrict__ summed) {
    int tid = blockIdx.x * blockDim.x + threadIdx.x;   // E*16 threads
    int e = tid >> 4;
    int f = (tid & 15) * 4;
    if (e < N_EDGES) {
        int src = (int)ei[e];
        int dst = (int)ei[N_EDGES + e];
        const float4 v = *(const float4*)(feat + (size_t)src * 64 + f);
        float* o = summed + (size_t)dst * 64 + f;
        atomicAdd(o + 0, v.x);
        atomicAdd(o + 1, v.y);
        atomicAdd(o + 2, v.z);
        atomicAdd(o + 3, v.w);
    }
}

// ---------------------------------------------------------------------------
// Fused SAGE linear layer via V_WMMA_F32_16X16X4_F32:
//   out[16x16 tile] = (summed*rdeg) @ Wl + xin @ Wr + bias   (optional ReLU)
// One wave32 per 16x16 output tile; K=64 -> 16 WMMAs per term, 32 chained.
//
// NOUT is a compile-time constant so all loop loads/stores use immediate
// byte offsets (no per-iteration 64-bit address arithmetic); the half-lane
// variant components are folded into base pointers once.
//
// Fragment layouts (ISA 7.12.2):
//   A 16x4 f32 : lanes 0-15 -> M=lane, K={0,1}; lanes 16-31 -> K={2,3}
//   B 4x16 f32 : lanes 0-15 -> N=lane, K rows {0,1}; lanes 16-31 -> rows {2,3}
//   C/D 16x16  : v[j] -> M = j + 8*(lane>=16), N = lane&15
// ---------------------------------------------------------------------------
template <int NOUT, bool RELU>
__global__ void sage_gemm_wmma(const float* __restrict__ summed, // [N, 64]
                               const float* __restrict__ rdeg,   // [N]
                               const float* __restrict__ xin,    // [N, 64]
                               const float* __restrict__ Wl,     // [64, NOUT]
                               const float* __restrict__ Wr,     // [64, NOUT]
                               const float* __restrict__ bias,   // [NOUT]
                               float* __restrict__ out) {        // [N, NOUT]
    const int lane = threadIdx.x;      // 0..31, wave32
    const int half = lane >> 4;        // 0 or 1
    const int lid  = lane & 15;
    const int row0 = blockIdx.x * 16;  // node tile
    const int col0 = blockIdx.y * 16;  // output-feature tile
    const int row  = row0 + lid;
    const int col  = col0 + lid;

    const float r = rdeg[row];

    // Base pointers with all lane-variant components folded in; loop indices
    // become pure compile-time immediate offsets.
    const float* __restrict__ arow = summed + (size_t)row * 64 + half * 2;
    const float* __restrict__ xrow = xin    + (size_t)row * 64 + half * 2;
    const float* __restrict__ wl   = Wl + col + half * 2 * NOUT;
    const float* __restrict__ wr   = Wr + col + half * 2 * NOUT;

    v8f c = {};

    // ---- (summed * rdeg) @ Wl : 16 chained K=4 WMMAs over K=64 ----
#pragma unroll
    for (int k0 = 0; k0 < 16; ++k0) {
        v2f a; a[0] = arow[k0 * 4] * r;       a[1] = arow[k0 * 4 + 1] * r;
        v2f b; b[0] = wl[(k0 * 4) * NOUT];    b[1] = wl[(k0 * 4 + 1) * NOUT];
        c = __builtin_amdgcn_wmma_f32_16x16x4_f32(
                /*neg_a=*/false, a, /*neg_b=*/false, b,
                /*c_mod=*/(short)0, c, /*reuse_a=*/false, /*reuse_b=*/false);
    }

    // ---- xin @ Wr : 16 more WMMAs into the same accumulator ----
#pragma unroll
    for (int k0 = 0; k0 < 16; ++k0) {
        v2f a; a[0] = xrow[k0 * 4];           a[1] = xrow[k0 * 4 + 1];
        v2f b; b[0] = wr[(k0 * 4) * NOUT];    b[1] = wr[(k0 * 4 + 1) * NOUT];
        c = __builtin_amdgcn_wmma_f32_16x16x4_f32(
                false, a, false, b, (short)0, c, false, false);
    }

    // ---- bias + optional ReLU + store (immediate row offsets) ----
    const float bb = bias[col];
    float* __restrict__ orow = out + (size_t)(row0 + half * 8) * NOUT + col;
#pragma unroll
    for (int j = 0; j < 8; ++j) {
        float val = c[j] + bb;
        if (RELU) val = fmaxf(val, 0.0f);
        orow[j * NOUT] = val;
    }
}

// ---------------------------------------------------------------------------
// Launch: full 2-layer GraphSAGE forward
// ---------------------------------------------------------------------------
extern "C" void kernel_launch(void* const* d_in, const int* in_sizes, int n_in,
                              void* d_out, int out_size, void* d_ws, size_t ws_size,
                              hipStream_t stream) {
    const float*     x    = (const float*)d_in[0];
    const long long* ei   = (const long long*)d_in[1];  // int64 [2, E]
    const float*     W1_l = (const float*)d_in[2];
    const float*     b1   = (const float*)d_in[3];
    const float*     W1_r = (const float*)d_in[4];
    const float*     W2_l = (const float*)d_in[5];
    const float*     b2   = (const float*)d_in[6];
    const float*     W2_r = (const float*)d_in[7];
    float*           out  = (float*)d_out;

    // Workspace layout (floats): deg[N] | rdeg[N] | summed[N*64] | h[N*64]
    float* deg    = (float*)d_ws;
    float* rdeg   = deg  + N_NODES;
    float* summed = rdeg + N_NODES;
    float* h      = summed + (size_t)N_NODES * D_HID;

    const int TB = 256;
    const int nSum = N_NODES * D_HID;

    // --- degree (shared by both layers) ---
    zero_f32<<<(N_NODES + TB - 1) / TB, TB, 0, stream>>>(deg, N_NODES);
    degree_kernel<<<(N_EDGES + TB - 1) / TB, TB, 0, stream>>>(ei, deg);
    rdeg_kernel<<<(N_NODES + TB - 1) / TB, TB, 0, stream>>>(deg, rdeg);

    // --- layer 1: aggregate x, then h = relu(mean@W1_l + x@W1_r + b1) ---
    zero_f32<<<(nSum + TB - 1) / TB, TB, 0, stream>>>(summed, nSum);
    {
        int threads = N_EDGES * 16;
        scatter_kernel<<<(threads + TB - 1) / TB, TB, 0, stream>>>(ei, x, summed);
    }
    {
        dim3 grid(N_NODES / 16, D_HID / 16);   // 6250 x 4 tiles
        sage_gemm_wmma<D_HID, true><<<grid, 32, 0, stream>>>(
            summed, rdeg, x, W1_l, W1_r, b1, h);
    }

    // --- layer 2: aggregate h, then out = mean@W2_l + h@W2_r + b2 ---
    zero_f32<<<(nSum + TB - 1) / TB, TB, 0, stream>>>(summed, nSum);
    {
        int threads = N_EDGES * 16;
        scatter_kernel<<<(threads + TB - 1) / TB, TB, 0, stream>>>(ei, h, summed);
    }
    {
        dim3 grid(N_NODES / 16, D_OUT / 16);   // 6250 x 2 tiles
        sage_gemm_wmma<D_OUT, false><<<grid, 32, 0, stream>>>(
            summed, rdeg, h, W2_l, W2_r, b2, out);
    }
}